// GraphConvPoolNNCOLLAB_18270790877378
// MI455X (gfx1250) — compile-verified
//
#include <hip/hip_runtime.h>
#include <hip/hip_bf16.h>

// ---------------------------------------------------------------------------
// GraphConvPool (GCN x2 -> topK pool -> GCN x2 -> mean pool -> MLP head)
// CDNA5 / gfx1250 implementation. fp32 WMMA (V_WMMA_F32_16X16X4_F32) for all
// node-feature GEMMs; atomic-f32 scatter for edge aggregation (L2-resident);
// radix-select for top-K (selection set is order-invariant for this model).
// ---------------------------------------------------------------------------

#define NN  50000
#define EE  800000
#define DD  128
#define HH  128
#define BB  64
#define CC  3
#define KK  (NN / 2)

typedef __attribute__((ext_vector_type(2))) float v2f;
typedef __attribute__((ext_vector_type(8))) float v8f;

// ------------------------------ GEMM (WMMA) --------------------------------
// Out[M x 128] = A[M x 128] @ W[128 x 128], fp32 WMMA 16x16x4.
// One wave computes a 16-row x 128-col slab: 8 accumulators, 32 k-steps.
__global__ __launch_bounds__(256) void k_gemm128_wmma(
    const float* __restrict__ A, const float* __restrict__ W,
    float* __restrict__ Out, int M)
{
    const int wave = threadIdx.x >> 5;
    const int lane = threadIdx.x & 31;
    const int m0 = (blockIdx.x * 8 + wave) * 16;
    if (m0 >= M) return;

    const int half = lane >> 4;   // 0: K=k,k+1   1: K=k+2,k+3
    const int l15  = lane & 15;
    const int koff = half * 2;

    int arow = m0 + l15;
    if (arow >= M) arow = M - 1;  // clamp; stores are guarded

    v8f acc[8];
    const v8f vzero = {0.f,0.f,0.f,0.f,0.f,0.f,0.f,0.f};
    #pragma unroll
    for (int t = 0; t < 8; ++t) acc[t] = vzero;

    for (int kk = 0; kk < 128; kk += 4) {
        // A fragment: 16x4 tile.  lane(0-15): rows m0..m0+15, K=kk,kk+1
        //                         lane(16-31): same rows,     K=kk+2,kk+3
        const float* ap = A + (size_t)arow * 128 + kk + koff;
        v2f a; a.x = ap[0]; a.y = ap[1];
        #pragma unroll
        for (int t = 0; t < 8; ++t) {
            // B fragment: rows kk..kk+3 of W, cols t*16..t*16+15
            const float* bp = W + (size_t)(kk + koff) * 128 + t * 16 + l15;
            v2f b; b.x = bp[0]; b.y = bp[128];
            acc[t] = __builtin_amdgcn_wmma_f32_16x16x4_f32(
                false, a, false, b, (short)0, acc[t], false, false);
        }
    }

    #pragma unroll
    for (int t = 0; t < 8; ++t) {
        #pragma unroll
        for (int v = 0; v < 8; ++v) {
            int row = m0 + v + half * 8;
            if (row < M) Out[(size_t)row * 128 + t * 16 + l15] = acc[t][v];
        }
    }
}

// ------------------------------ utilities ----------------------------------
__global__ void k_fill_f(float* p, float v, int n) {
    int i = blockIdx.x * blockDim.x + threadIdx.x;
    if (i < n) p[i] = v;
}
__global__ void k_fill_i(int* p, int v, int n) {
    int i = blockIdx.x * blockDim.x + threadIdx.x;
    if (i < n) p[i] = v;
}
__global__ void k_rsqrt_inplace(float* p, int n) {
    int i = blockIdx.x * blockDim.x + threadIdx.x;
    if (i < n) p[i] = rsqrtf(p[i]);
}
__global__ void k_pnorm(const float* __restrict__ p, float* out) {
    __shared__ float sm[128];
    int f = threadIdx.x;
    float v = p[f];
    sm[f] = v * v;
    __syncthreads();
    for (int s = 64; s > 0; s >>= 1) {
        if (f < s) sm[f] += sm[f + s];
        __syncthreads();
    }
    if (f == 0) out[0] = rsqrtf(sm[0]);
}

// ------------------------- degree / aggregation ----------------------------
__global__ void k_deg_accum(const int* __restrict__ ei, float* __restrict__ deg, int e) {
    int i = blockIdx.x * blockDim.x + threadIdx.x;
    if (i < e) atomicAdd(&deg[ei[2 * i + 1]], 1.0f);
}
__global__ void k_degP_accum(const int* __restrict__ ei, const int* __restrict__ nidx,
                             float* __restrict__ deg, int e) {
    int i = blockIdx.x * blockDim.x + threadIdx.x;
    if (i >= e) return;
    int ns = nidx[ei[2 * i]], nd = nidx[ei[2 * i + 1]];
    if (ns >= 0 && nd >= 0) atomicAdd(&deg[nd], 1.0f);
}
// wave-per-edge scatter: agg[dst] += dinv[src]*dinv[dst]*h[src]
__global__ void k_scatter(const float* __restrict__ h, const int* __restrict__ ei,
                          const float* __restrict__ dinv, float* __restrict__ agg, int e) {
    int tid = blockIdx.x * blockDim.x + threadIdx.x;
    int ed = tid >> 5, lane = tid & 31;
    if (ed >= e) return;
    int s = ei[2 * ed], d = ei[2 * ed + 1];
    float c = dinv[s] * dinv[d];
    const float* hs = h + (size_t)s * 128;
    float* ad = agg + (size_t)d * 128;
    #pragma unroll
    for (int j = 0; j < 4; ++j) {
        int f = lane + 32 * j;
        atomicAdd(&ad[f], c * hs[f]);
    }
}
__global__ void k_scatterP(const float* __restrict__ h, const int* __restrict__ ei,
                           const int* __restrict__ nidx, const float* __restrict__ dinv,
                           float* __restrict__ agg, int e) {
    int tid = blockIdx.x * blockDim.x + threadIdx.x;
    int ed = tid >> 5, lane = tid & 31;
    if (ed >= e) return;
    int ns = nidx[ei[2 * ed]], nd = nidx[ei[2 * ed + 1]];
    if (ns < 0 || nd < 0) return;
    float c = dinv[ns] * dinv[nd];
    const float* hs = h + (size_t)ns * 128;
    float* ad = agg + (size_t)nd * 128;
    #pragma unroll
    for (int j = 0; j < 4; ++j) {
        int f = lane + 32 * j;
        atomicAdd(&ad[f], c * hs[f]);
    }
}
// h := relu(agg + dinv^2 * h + b)    (in place over h)
__global__ void k_finalize(float* __restrict__ hio, const float* __restrict__ agg,
                           const float* __restrict__ dinv, const float* __restrict__ b, int n) {
    int i = blockIdx.x * blockDim.x + threadIdx.x;
    if (i >= n * 128) return;
    int node = i >> 7, f = i & 127;
    float di = dinv[node];
    float v = agg[i] + di * di * hio[i] + b[f];
    hio[i] = fmaxf(v, 0.0f);
}

// ------------------------------ score / topK -------------------------------
__global__ void k_score(const float* __restrict__ h, const float* __restrict__ p,
                        const float* __restrict__ pninv, float* __restrict__ score, int n) {
    int tid = blockIdx.x * blockDim.x + threadIdx.x;
    int node = tid >> 5, lane = tid & 31;
    if (node >= n) return;
    const float* hr = h + (size_t)node * 128;
    float s = 0.0f;
    #pragma unroll
    for (int j = 0; j < 4; ++j) s += hr[lane + 32 * j] * p[lane + 32 * j];
    #pragma unroll
    for (int o = 16; o > 0; o >>= 1) s += __shfl_xor(s, o, 32);
    if (lane == 0) score[node] = s * pninv[0];
}

__device__ __forceinline__ unsigned f2key(float s) {
    unsigned u = __float_as_uint(s);
    return (u & 0x80000000u) ? ~u : (u | 0x80000000u);  // monotonic
}
// ctrl: [0]=prefix(u32) [1]=remaining [2]=eqCnt [3]=selCnt [4]=T(u32) [5]=needEq
__global__ void k_topk_init(int* ctrl, int k) {
    ((unsigned*)ctrl)[0] = 0u; ctrl[1] = k; ctrl[2] = 0; ctrl[3] = 0;
    ((unsigned*)ctrl)[4] = 0u; ctrl[5] = 0;
}
__global__ void k_hist_zero(int* hist) { hist[threadIdx.x] = 0; }
__global__ void k_hist(const float* __restrict__ score, const int* __restrict__ ctrl,
                       int* __restrict__ hist, int shift, int n) {
    int i = blockIdx.x * blockDim.x + threadIdx.x;
    if (i >= n) return;
    unsigned u = f2key(score[i]);
    unsigned pm = (shift == 24) ? 0u : (0xFFFFFFFFu << (shift + 8));
    if ((u & pm) == ((const unsigned*)ctrl)[0])
        atomicAdd(&hist[(u >> shift) & 255], 1);
}
__global__ void k_scan(int* ctrl, const int* __restrict__ hist, int shift) {
    unsigned pref = ((unsigned*)ctrl)[0];
    int rem = ctrl[1];
    int cum = 0;
    for (int b = 255; b >= 0; --b) {
        int c = hist[b];
        if (cum + c >= rem) {
            pref |= ((unsigned)b) << shift;
            ((unsigned*)ctrl)[0] = pref;
            ctrl[1] = rem - cum;
            if (shift == 0) { ((unsigned*)ctrl)[4] = pref; ctrl[5] = rem - cum; }
            return;
        }
        cum += c;
    }
}
__global__ void k_select(const float* __restrict__ score, const int* __restrict__ batch,
                         int* ctrl, int* __restrict__ new_idx, int* __restrict__ old_of_new,
                         int* __restrict__ batch_p, int n) {
    int i = blockIdx.x * blockDim.x + threadIdx.x;
    if (i >= n) return;
    unsigned u = f2key(score[i]);
    unsigned T = ((unsigned*)ctrl)[4];
    bool sel = false;
    if (u > T) sel = true;
    else if (u == T) {
        int pos = atomicAdd(&ctrl[2], 1);
        if (pos < ctrl[5]) sel = true;
    }
    if (sel) {
        int j = atomicAdd(&ctrl[3], 1);
        new_idx[i] = j;
        old_of_new[j] = i;
        batch_p[j] = batch[i];
    }
}
// hp[j] = h2[old_of_new[j]] * tanh(score[old_of_new[j]])
__global__ void k_gather_hp(const float* __restrict__ h2, const float* __restrict__ score,
                            const int* __restrict__ old_of_new, float* __restrict__ hp, int k) {
    int tid = blockIdx.x * blockDim.x + threadIdx.x;
    int j = tid >> 5, lane = tid & 31;
    if (j >= k) return;
    int n = old_of_new[j];
    float g = tanhf(score[n]);
    const float* src = h2 + (size_t)n * 128;
    float* dst = hp + (size_t)j * 128;
    #pragma unroll
    for (int q = 0; q < 4; ++q) {
        int f = lane + 32 * q;
        dst[f] = src[f] * g;
    }
}

// ------------------------------ pooling / head -----------------------------
__global__ void k_pool(const float* __restrict__ h, const int* __restrict__ batch_p,
                       float* __restrict__ sums, float* __restrict__ cnt, int k) {
    int tid = blockIdx.x * blockDim.x + threadIdx.x;
    int j = tid >> 5, lane = tid & 31;
    if (j >= k) return;
    int b = batch_p[j];
    const float* hr = h + (size_t)j * 128;
    float* sr = sums + (size_t)b * 128;
    #pragma unroll
    for (int q = 0; q < 4; ++q) {
        int f = lane + 32 * q;
        atomicAdd(&sr[f], hr[f]);
    }
    if (lane == 0) atomicAdd(&cnt[b], 1.0f);
}
__global__ void k_head(const float* __restrict__ sums, const float* __restrict__ cnt,
                       const float* __restrict__ fcW1, const float* __restrict__ fcb1,
                       const float* __restrict__ fcW2, const float* __restrict__ fcb2,
                       float* __restrict__ out) {
    __shared__ float mean[128];
    __shared__ float g[128];
    __shared__ float logit[3];
    int b = blockIdx.x, f = threadIdx.x;
    float c = fmaxf(cnt[b], 1.0f);
    mean[f] = sums[(size_t)b * 128 + f] / c;
    __syncthreads();
    float acc = fcb1[f];
    for (int q = 0; q < 128; ++q) acc += mean[q] * fcW1[q * 128 + f];
    g[f] = fmaxf(acc, 0.0f);
    __syncthreads();
    if (f < 3) {
        float a = fcb2[f];
        for (int q = 0; q < 128; ++q) a += g[q] * fcW2[q * 3 + f];
        logit[f] = a;
    }
    __syncthreads();
    if (f == 0) {
        float m = fmaxf(logit[0], fmaxf(logit[1], logit[2]));
        float s = expf(logit[0] - m) + expf(logit[1] - m) + expf(logit[2] - m);
        float l = logf(s) + m;
        out[b * 3 + 0] = logit[0] - l;
        out[b * 3 + 1] = logit[1] - l;
        out[b * 3 + 2] = logit[2] - l;
    }
}

// ------------------------------- launcher ----------------------------------
static inline int cdiv(long long a, long long b) { return (int)((a + b - 1) / b); }

extern "C" void kernel_launch(void* const* d_in, const int* in_sizes, int n_in,
                              void* d_out, int out_size, void* d_ws, size_t ws_size,
                              hipStream_t stream) {
    const float* x     = (const float*)d_in[0];
    const int*   ei    = (const int*)  d_in[1];
    const int*   batch = (const int*)  d_in[2];
    const float* W1    = (const float*)d_in[3];
    const float* b1    = (const float*)d_in[4];
    const float* W2    = (const float*)d_in[5];
    const float* b2    = (const float*)d_in[6];
    const float* p     = (const float*)d_in[7];
    const float* W3    = (const float*)d_in[8];
    const float* b3    = (const float*)d_in[9];
    const float* W4    = (const float*)d_in[10];
    const float* b4    = (const float*)d_in[11];
    const float* fcW1  = (const float*)d_in[12];
    const float* fcb1  = (const float*)d_in[13];
    const float* fcW2  = (const float*)d_in[14];
    const float* fcb2  = (const float*)d_in[15];
    float* out = (float*)d_out;

    // workspace layout
    char* ws = (char*)d_ws;
    size_t off = 0;
    auto alloc = [&](size_t bytes) -> char* {
        char* r = ws + off;
        off += (bytes + 255) & ~(size_t)255;
        return r;
    };
    const size_t BIG = (size_t)NN * 128 * sizeof(float);
    float* buf1 = (float*)alloc(BIG);  // t1/h1 ; later: hp (rows 0..K-1), t3/h3 (rows K..2K-1)
    float* buf2 = (float*)alloc(BIG);  // t2/h2 ; later: aggP (rows 0..K-1), t4/h4 (rows K..2K-1)
    float* buf3 = (float*)alloc(BIG);  // agg conv1/2 ; later agg4 (rows 0..K-1)
    float* dinv   = (float*)alloc((size_t)NN * sizeof(float));
    float* dinvP  = (float*)alloc((size_t)KK * sizeof(float));
    float* score  = (float*)alloc((size_t)NN * sizeof(float));
    int*   newidx = (int*)  alloc((size_t)NN * sizeof(int));
    int*   old_of = (int*)  alloc((size_t)KK * sizeof(int));
    int*   batchp = (int*)  alloc((size_t)KK * sizeof(int));
    int*   hist   = (int*)  alloc(256 * sizeof(int));
    int*   ctrl   = (int*)  alloc(8 * sizeof(int));
    float* pninv  = (float*)alloc(sizeof(float));
    float* sums   = (float*)alloc((size_t)BB * 128 * sizeof(float));
    float* cnt    = (float*)alloc((size_t)BB * sizeof(float));

    float* hp = buf1;                       // K x 128
    float* t3 = buf1 + (size_t)KK * 128;    // K x 128 (becomes h3)
    float* aggP = buf2;                     // K x 128
    float* t4 = buf2 + (size_t)KK * 128;    // K x 128 (becomes h4)
    float* agg4 = buf3;                     // K x 128

    const int TB = 256;
    const int gN      = cdiv(NN, TB);
    const int gK      = cdiv(KK, TB);
    const int gE      = cdiv(EE, TB);
    const int gEw     = cdiv((long long)EE * 32, TB);   // wave per edge
    const int gNw     = cdiv((long long)NN * 32, TB);   // wave per node
    const int gKw     = cdiv((long long)KK * 32, TB);
    const int gNfeat  = cdiv((long long)NN * 128, TB);
    const int gKfeat  = cdiv((long long)KK * 128, TB);
    const int gGemmN  = cdiv(cdiv(NN, 16), 8);
    const int gGemmK  = cdiv(cdiv(KK, 16), 8);

    // ---- shared precompute ----
    k_pnorm<<<1, 128, 0, stream>>>(p, pninv);
    k_fill_f<<<gN, TB, 0, stream>>>(dinv, 1.0f, NN);
    k_deg_accum<<<gE, TB, 0, stream>>>(ei, dinv, EE);
    k_rsqrt_inplace<<<gN, TB, 0, stream>>>(dinv, NN);

    // ---- conv1: h1 = relu(gcn(x, W1, b1)) ----
    k_gemm128_wmma<<<gGemmN, TB, 0, stream>>>(x, W1, buf1, NN);
    k_fill_f<<<gNfeat, TB, 0, stream>>>(buf3, 0.0f, NN * 128);
    k_scatter<<<gEw, TB, 0, stream>>>(buf1, ei, dinv, buf3, EE);
    k_finalize<<<gNfeat, TB, 0, stream>>>(buf1, buf3, dinv, b1, NN);

    // ---- conv2: h2 = relu(gcn(h1, W2, b2)) ----
    k_gemm128_wmma<<<gGemmN, TB, 0, stream>>>(buf1, W2, buf2, NN);
    k_fill_f<<<gNfeat, TB, 0, stream>>>(buf3, 0.0f, NN * 128);
    k_scatter<<<gEw, TB, 0, stream>>>(buf2, ei, dinv, buf3, EE);
    k_finalize<<<gNfeat, TB, 0, stream>>>(buf2, buf3, dinv, b2, NN);

    // ---- score + radix-select top K ----
    k_score<<<gNw, TB, 0, stream>>>(buf2, p, pninv, score, NN);
    k_topk_init<<<1, 1, 0, stream>>>(ctrl, KK);
    for (int shift = 24; shift >= 0; shift -= 8) {
        k_hist_zero<<<1, 256, 0, stream>>>(hist);
        k_hist<<<gN, TB, 0, stream>>>(score, ctrl, hist, shift, NN);
        k_scan<<<1, 1, 0, stream>>>(ctrl, hist, shift);
    }
    k_fill_i<<<gN, TB, 0, stream>>>(newidx, -1, NN);
    k_select<<<gN, TB, 0, stream>>>(score, batch, ctrl, newidx, old_of, batchp, NN);
    k_gather_hp<<<gKw, TB, 0, stream>>>(buf2, score, old_of, hp, KK);  // h2 dead after this

    // ---- pooled graph degrees ----
    k_fill_f<<<gK, TB, 0, stream>>>(dinvP, 1.0f, KK);
    k_degP_accum<<<gE, TB, 0, stream>>>(ei, newidx, dinvP, EE);
    k_rsqrt_inplace<<<gK, TB, 0, stream>>>(dinvP, KK);

    // ---- conv3: h3 = relu(gcn(hp, W3, b3)) ----
    k_gemm128_wmma<<<gGemmK, TB, 0, stream>>>(hp, W3, t3, KK);
    k_fill_f<<<gKfeat, TB, 0, stream>>>(aggP, 0.0f, KK * 128);
    k_scatterP<<<gEw, TB, 0, stream>>>(t3, ei, newidx, dinvP, aggP, EE);
    k_finalize<<<gKfeat, TB, 0, stream>>>(t3, aggP, dinvP, b3, KK);    // t3 -> h3

    // ---- conv4: h4 = relu(gcn(h3, W4, b4)) ----
    k_gemm128_wmma<<<gGemmK, TB, 0, stream>>>(t3, W4, t4, KK);
    k_fill_f<<<gKfeat, TB, 0, stream>>>(agg4, 0.0f, KK * 128);
    k_scatterP<<<gEw, TB, 0, stream>>>(t4, ei, newidx, dinvP, agg4, EE);
    k_finalize<<<gKfeat, TB, 0, stream>>>(t4, agg4, dinvP, b4, KK);    // t4 -> h4

    // ---- mean pool + MLP head + log_softmax ----
    k_fill_f<<<cdiv(BB * 128, TB), TB, 0, stream>>>(sums, 0.0f, BB * 128);
    k_fill_f<<<1, BB, 0, stream>>>(cnt, 0.0f, BB);
    k_pool<<<gKw, TB, 0, stream>>>(t4, batchp, sums, cnt, KK);
    k_head<<<BB, 128, 0, stream>>>(sums, cnt, fcW1, fcb1, fcW2, fcb2, out);

    (void)in_sizes; (void)n_in; (void)out_size; (void)ws_size;
}